// GcnNet_64991445123398
// MI455X (gfx1250) — compile-verified
//
#include <hip/hip_runtime.h>
#include <hip/hip_bf16.h>

#define N_NODES  100000
#define N_EDGES  1600000
#define N_GRAPHS 2048
#define D_IN     128
#define HDIM     64
#define N_CLASS  10

typedef float v2f __attribute__((ext_vector_type(2)));
typedef float v8f __attribute__((ext_vector_type(8)));

// ---------------- degree / norm ----------------

__global__ void deg_init_kernel(float* deg, int n) {
    int t = blockIdx.x * blockDim.x + threadIdx.x;
    if (t < n) deg[t] = 1.0f;   // self loop
}

__global__ void deg_count_kernel(const int* __restrict__ ei, float* deg, int e) {
    int t = blockIdx.x * blockDim.x + threadIdx.x;
    if (t < e) {
        int dst = ei[N_EDGES + t];
        atomicAdd(&deg[dst], 1.0f);
    }
}

__global__ void dinv_kernel(float* deg, int n) {
    int t = blockIdx.x * blockDim.x + threadIdx.x;
    if (t < n) deg[t] = rsqrtf(deg[t]);   // deg >= 1 always
}

// ---------------- dense GEMM via V_WMMA_F32_16X16X4_F32 ----------------
// One wave32 computes a full 16x64 output strip (4 accumulator tiles),
// reusing each A fragment across all 4 column tiles. W is staged in LDS.
// Epilogue writes h AND initializes agg = bias + dinv^2 * h (fused self-loop).

template <int D>
__global__ void gemm_wmma_f32_kernel(const float* __restrict__ A,
                                     const float* __restrict__ W,
                                     float* __restrict__ Hout,
                                     float* __restrict__ agg,
                                     const float* __restrict__ dinv,
                                     const float* __restrict__ bias) {
    __shared__ float sW[D * HDIM];

    // cooperative load of W into LDS (float4 vectorized)
    {
        const int n4 = (D * HDIM) / 4;
        const float4* Wg = reinterpret_cast<const float4*>(W);
        float4* Ws = reinterpret_cast<float4*>(sW);
        for (int i = threadIdx.x; i < n4; i += blockDim.x) Ws[i] = Wg[i];
    }
    __syncthreads();

    const int wave = blockIdx.x * (blockDim.x >> 5) + (threadIdx.x >> 5);
    const int lane = threadIdx.x & 31;
    const int half = lane >> 4;             // 0 or 1
    const int lp   = lane & 15;

    const int row_base = wave * 16;         // grid sized exactly: always valid

    v8f cacc[4] = {v8f{}, v8f{}, v8f{}, v8f{}};

    const float* arow = A + (size_t)(row_base + lp) * D;

#pragma unroll
    for (int k = 0; k < D; k += 4) {
        // A 16x4 tile: lane lp holds row M=lp; K = 2*half + {0,1}
        v2f a = *reinterpret_cast<const v2f*>(arow + k + 2 * half);
        // B 4x16 tiles from LDS: lane lp holds col N; K = 2*half + {0,1}
        const float* w0 = sW + (k + 2 * half) * HDIM + lp;
        const float* w1 = w0 + HDIM;
#pragma unroll
        for (int ct = 0; ct < 4; ++ct) {
            v2f b;
            b.x = w0[ct * 16];
            b.y = w1[ct * 16];
            cacc[ct] = __builtin_amdgcn_wmma_f32_16x16x4_f32(
                /*neg_a=*/false, a, /*neg_b=*/false, b,
                /*c_mod=*/(short)0, cacc[ct], /*reuse_a=*/false, /*reuse_b=*/false);
        }
    }

    // C/D layout: VGPR v, lanes 0-15 -> M=v, lanes 16-31 -> M=v+8; N=lp
    const int row0 = row_base + 8 * half;
#pragma unroll
    for (int v = 0; v < 8; ++v) {
        const int row = row0 + v;
        const float di = dinv[row];
        const float d2 = di * di;
        const size_t rbase = (size_t)row * HDIM;
#pragma unroll
        for (int ct = 0; ct < 4; ++ct) {
            const int col = ct * 16 + lp;
            const float val = cacc[ct][v];
            Hout[rbase + col] = val;
            agg[rbase + col] = bias[col] + d2 * val;
        }
    }
}

// ---------------- aggregation ----------------

// One wave32 per edge: lane l handles columns 2l, 2l+1.
__global__ void edge_agg_kernel(const int* __restrict__ ei,
                                const float* __restrict__ dinv,
                                const float* __restrict__ h,
                                float* __restrict__ agg) {
    int t = blockIdx.x * blockDim.x + threadIdx.x;
    int e = t >> 5;
    int lane = threadIdx.x & 31;
    if (e >= N_EDGES) return;
    int src = ei[e];
    int dst = ei[N_EDGES + e];
    float w = dinv[src] * dinv[dst];
    v2f hv = *reinterpret_cast<const v2f*>(h + (size_t)src * HDIM + 2 * lane);
    float* ap = agg + (size_t)dst * HDIM + 2 * lane;
    atomicAdd(ap + 0, w * hv.x);
    atomicAdd(ap + 1, w * hv.y);
}

__global__ void tanh_kernel(float* buf, int n) {
    int t = blockIdx.x * blockDim.x + threadIdx.x;
    if (t < n) buf[t] = tanhf(buf[t]);
}

// ---------------- pooling + head ----------------

__global__ void zero_kernel(float* buf, int n) {
    int t = blockIdx.x * blockDim.x + threadIdx.x;
    if (t < n) buf[t] = 0.0f;
}

__global__ void pool_kernel(const float* __restrict__ act,
                            const int* __restrict__ batch,
                            float* __restrict__ pooled, int n) {
    int t = blockIdx.x * blockDim.x + threadIdx.x;
    if (t >= n) return;
    int i = t >> 6;
    int j = t & 63;
    atomicAdd(&pooled[batch[i] * HDIM + j], act[t]);
}

__global__ void head_kernel(const float* __restrict__ pooled,
                            const float* __restrict__ Wf,
                            const float* __restrict__ bf,
                            float* __restrict__ out, int n) {
    int t = blockIdx.x * blockDim.x + threadIdx.x;
    if (t >= n) return;
    int g = t / N_CLASS;
    int c = t - g * N_CLASS;
    const float* p = pooled + g * HDIM;
    float acc = bf[c];
#pragma unroll
    for (int j = 0; j < HDIM; ++j) {
        acc += p[j] * Wf[j * N_CLASS + c];
    }
    out[t] = tanhf(acc);
}

// ---------------- launch ----------------

static inline int cdiv(long long a, int b) { return (int)((a + b - 1) / b); }

extern "C" void kernel_launch(void* const* d_in, const int* in_sizes, int n_in,
                              void* d_out, int out_size, void* d_ws, size_t ws_size,
                              hipStream_t stream) {
    const float* x     = (const float*)d_in[0];
    const int*   ei    = (const int*)  d_in[1];   // [2, E]
    const int*   batch = (const int*)  d_in[2];   // [N]
    const float* W1    = (const float*)d_in[3];
    const float* b1    = (const float*)d_in[4];
    const float* W2    = (const float*)d_in[5];
    const float* b2    = (const float*)d_in[6];
    const float* W3    = (const float*)d_in[7];
    const float* b3    = (const float*)d_in[8];
    const float* Wf    = (const float*)d_in[9];
    const float* bf    = (const float*)d_in[10];
    float* out = (float*)d_out;

    const size_t NH = (size_t)N_NODES * HDIM;     // 6.4M floats
    float* ws     = (float*)d_ws;
    float* dinv   = ws;                           // N
    float* h      = dinv + N_NODES;               // N*H
    float* aggA   = h + NH;                       // N*H
    float* aggB   = aggA + NH;                    // N*H
    float* pooled = aggB + NH;                    // G*H

    const int BLK = 256;

    // normalization coefficients
    deg_init_kernel<<<cdiv(N_NODES, BLK), BLK, 0, stream>>>(dinv, N_NODES);
    deg_count_kernel<<<cdiv(N_EDGES, BLK), BLK, 0, stream>>>(ei, dinv, N_EDGES);
    dinv_kernel<<<cdiv(N_NODES, BLK), BLK, 0, stream>>>(dinv, N_NODES);

    // GEMM grid: one wave per 16-row strip; 10 waves/block -> exactly 625 blocks
    const int GEMM_BLK   = 320;                       // 10 wave32
    const int gemm_waves = N_NODES / 16;              // 6250
    const int gemm_grid  = gemm_waves / (GEMM_BLK / 32);  // 625

    const long long edge_threads = (long long)N_EDGES * 32;

    layerwise:
    {
        // layer 1 (D = 128)
        gemm_wmma_f32_kernel<D_IN><<<gemm_grid, GEMM_BLK, 0, stream>>>(x, W1, h, aggA, dinv, b1);
        edge_agg_kernel<<<cdiv(edge_threads, BLK), BLK, 0, stream>>>(ei, dinv, h, aggA);
        tanh_kernel<<<cdiv((long long)NH, BLK), BLK, 0, stream>>>(aggA, (int)NH);

        // layer 2 (D = 64)
        gemm_wmma_f32_kernel<HDIM><<<gemm_grid, GEMM_BLK, 0, stream>>>(aggA, W2, h, aggB, dinv, b2);
        edge_agg_kernel<<<cdiv(edge_threads, BLK), BLK, 0, stream>>>(ei, dinv, h, aggB);
        tanh_kernel<<<cdiv((long long)NH, BLK), BLK, 0, stream>>>(aggB, (int)NH);

        // layer 3 (D = 64)
        gemm_wmma_f32_kernel<HDIM><<<gemm_grid, GEMM_BLK, 0, stream>>>(aggB, W3, h, aggA, dinv, b3);
        edge_agg_kernel<<<cdiv(edge_threads, BLK), BLK, 0, stream>>>(ei, dinv, h, aggA);
        tanh_kernel<<<cdiv((long long)NH, BLK), BLK, 0, stream>>>(aggA, (int)NH);
    }

    const int GP = N_GRAPHS * HDIM;
    zero_kernel<<<cdiv(GP, BLK), BLK, 0, stream>>>(pooled, GP);
    pool_kernel<<<cdiv((long long)NH, BLK), BLK, 0, stream>>>(aggA, batch, pooled, (int)NH);
    head_kernel<<<cdiv(N_GRAPHS * N_CLASS, BLK), BLK, 0, stream>>>(pooled, Wf, bf, out,
                                                                   N_GRAPHS * N_CLASS);
}